// LV_NeuralODEModel_21629455302761
// MI455X (gfx1250) — compile-verified
//
#include <hip/hip_runtime.h>
#include <hip/hip_bf16.h>

// Neural ODE RK4, B=131072, SEQ=64, PRED=32, D=2, H=256, N=96 steps.
// One wave32 handles 16 batch elements. Matmul1 (incl. bias) runs on the XDL
// pipe via V_WMMA_F32_16X16X4_F32 (fp32 matrix op -> no precision loss over
// 95 chained RK4 steps). tanh + matmul2 (H=256 -> D=2) run on VALU/TRANS,
// with W2 coefficients broadcast from LDS and a single shfl_xor(16) reduction.
// NOTE: deliberately no register cap — capping at 256 VGPRs makes the
// allocator spill to scratch inside the hot loop. The
// uncapped build holds ~450 VGPRs (2 waves/SIMD) with zero spills; the
// s_set_vgpr_msb SALU overhead co-issues with VALU.

typedef __attribute__((ext_vector_type(2))) float v2f;
typedef __attribute__((ext_vector_type(8))) float v8f;

__device__ __forceinline__ float fast_tanh(float x) {
#if __has_builtin(__builtin_amdgcn_tanhf)
  return __builtin_amdgcn_tanhf(x);            // v_tanh_f32 (TRANS pipe)
#else
  // tanh(x) = 1 - 2/(exp(2x)+1);  exp(2x) = exp2(x * 2*log2(e))
  float e = __builtin_amdgcn_exp2f(x * 2.885390081777926815f);
  return fmaf(-2.0f, __builtin_amdgcn_rcpf(e + 1.0f), 1.0f);
#endif
}

// One evaluation of f(y) = tanh(y@W1 + b1)@W2 + b2 for 16 batch elements
// (one per lane-pair n / n+16). Returns (f0,f1) valid on ALL lanes.
__device__ __forceinline__ float2 f_eval(float2 y, bool hi,
                                         const float4* __restrict__ w2s,
                                         const v2f* af,
                                         float b20, float b21) {
  // B fragment of Y_aug^T (4x16):
  //  VGPR0: lanes0-15 = y0 (K=0 row), lanes16-31 = 1.0 (K=2 row -> bias)
  //  VGPR1: lanes0-15 = y1 (K=1 row), lanes16-31 = 0.0 (K=3 row -> pad)
  v2f b;
  b.x = hi ? 1.0f : y.x;
  b.y = hi ? 0.0f : y.y;

  float acc0 = 0.0f, acc1 = 0.0f;
  const int wq = hi ? 4 : 0;  // float4 offset for this half's hidden rows

#pragma unroll
  for (int t = 0; t < 16; ++t) {
    v8f c = {};
    // H^T tile t (rows = hidden 16t..16t+15, cols = batch):
    // lane n holds H[16t + r + 8*hi][n] in d[r], r=0..7  (its own batch!)
    v8f d = __builtin_amdgcn_wmma_f32_16x16x4_f32(
        /*neg_a=*/false, af[t], /*neg_b=*/false, b,
        /*c_mod=*/(short)0, c, /*reuse_a=*/false, /*reuse_b=*/false);

    const int qb = t * 8 + wq;  // w2s is float4 = {W2[h][0],W2[h][1],W2[h+1][0],W2[h+1][1]}
    float4 wA = w2s[qb + 0];
    float4 wB = w2s[qb + 1];
    float4 wC = w2s[qb + 2];
    float4 wD = w2s[qb + 3];

    float h;
    h = fast_tanh(d[0]); acc0 = fmaf(h, wA.x, acc0); acc1 = fmaf(h, wA.y, acc1);
    h = fast_tanh(d[1]); acc0 = fmaf(h, wA.z, acc0); acc1 = fmaf(h, wA.w, acc1);
    h = fast_tanh(d[2]); acc0 = fmaf(h, wB.x, acc0); acc1 = fmaf(h, wB.y, acc1);
    h = fast_tanh(d[3]); acc0 = fmaf(h, wB.z, acc0); acc1 = fmaf(h, wB.w, acc1);
    h = fast_tanh(d[4]); acc0 = fmaf(h, wC.x, acc0); acc1 = fmaf(h, wC.y, acc1);
    h = fast_tanh(d[5]); acc0 = fmaf(h, wC.z, acc0); acc1 = fmaf(h, wC.w, acc1);
    h = fast_tanh(d[6]); acc0 = fmaf(h, wD.x, acc0); acc1 = fmaf(h, wD.y, acc1);
    h = fast_tanh(d[7]); acc0 = fmaf(h, wD.z, acc0); acc1 = fmaf(h, wD.w, acc1);
  }

  // lane n and lane n+16 each hold half of batch n's dot products
  acc0 += __shfl_xor(acc0, 16, 32);
  acc1 += __shfl_xor(acc1, 16, 32);
  return make_float2(acc0 + b20, acc1 + b21);
}

__global__ __launch_bounds__(256)
void node_rk4_kernel(const float* __restrict__ x,
                     const float* __restrict__ W1,
                     const float* __restrict__ b1,
                     const float4* __restrict__ W2q,   // 256x2 fp32 -> 128 float4
                     const float* __restrict__ b2,
                     float2* __restrict__ out) {       // (B, 32, 2) -> float2[B*32]
  __shared__ float4 w2s[128];
  const int tid = threadIdx.x;
  if (tid < 128) w2s[tid] = W2q[tid];
  __syncthreads();

  const int lane = tid & 31;
  const int hl   = lane & 15;
  const bool hi  = lane >= 16;

  // A fragments of W1_aug^T (16x4 per tile), constant across all steps:
  //  VGPR0: lanes0-15 = W1[0][h] (K=0), lanes16-31 = b1[h] (K=2)
  //  VGPR1: lanes0-15 = W1[1][h] (K=1), lanes16-31 = 0     (K=3)
  v2f af[16];
#pragma unroll
  for (int t = 0; t < 16; ++t) {
    const int h = t * 16 + hl;
    af[t].x = hi ? b1[h]       : W1[h];
    af[t].y = hi ? 0.0f        : W1[256 + h];
  }
  const float b20 = b2[0];
  const float b21 = b2[1];

  const int gw = blockIdx.x * 8 + (tid >> 5);   // global wave id
  const int n  = gw * 16 + hl;                  // this lane's batch element

  // y0 = x[n, 0, :]  (x is (B, 64, 2) fp32 -> float2 stride 64)
  const float2* xin = (const float2*)x;
  float2 y = xin[(size_t)n * 64];

  float2* outn = out + (size_t)n * 32;

  const float dt  = 1.0f / 95.0f;
  const float hdt = 0.5f * dt;
  const float c6  = dt * (1.0f / 6.0f);

#pragma unroll 1
  for (int s = 1; s <= 95; ++s) {
    float2 k1 = f_eval(y, hi, w2s, af, b20, b21);
    float2 yt;
    yt.x = fmaf(hdt, k1.x, y.x);
    yt.y = fmaf(hdt, k1.y, y.y);
    float2 k2 = f_eval(yt, hi, w2s, af, b20, b21);
    yt.x = fmaf(hdt, k2.x, y.x);
    yt.y = fmaf(hdt, k2.y, y.y);
    float2 k3 = f_eval(yt, hi, w2s, af, b20, b21);
    yt.x = fmaf(dt, k3.x, y.x);
    yt.y = fmaf(dt, k3.y, y.y);
    float2 k4 = f_eval(yt, hi, w2s, af, b20, b21);

    y.x += c6 * (k1.x + 2.0f * (k2.x + k3.x) + k4.x);
    y.y += c6 * (k1.y + 2.0f * (k2.y + k3.y) + k4.y);

    if (s >= 64 && !hi) outn[s - 64] = y;   // last PRED=32 states
  }
}

extern "C" void kernel_launch(void* const* d_in, const int* in_sizes, int n_in,
                              void* d_out, int out_size, void* d_ws, size_t ws_size,
                              hipStream_t stream) {
  (void)in_sizes; (void)n_in; (void)out_size; (void)d_ws; (void)ws_size;
  const float*  x  = (const float*)d_in[0];   // (131072, 64, 2) f32
  const float*  W1 = (const float*)d_in[1];   // (2, 256) f32
  const float*  b1 = (const float*)d_in[2];   // (256,) f32
  const float4* W2 = (const float4*)d_in[3];  // (256, 2) f32 viewed as 128 float4
  const float*  b2 = (const float*)d_in[4];   // (2,) f32
  float2* out = (float2*)d_out;               // (131072, 32, 2) f32

  // 131072 batches / (8 waves * 16 batches per wave) = 1024 blocks of 256
  node_rk4_kernel<<<1024, 256, 0, stream>>>(x, W1, b1, W2, b2, out);
}